// PrepareDecoderInput_670014898636
// MI455X (gfx1250) — compile-verified
//
#include <hip/hip_runtime.h>
#include <hip/hip_bf16.h>

// ---- problem constants (from reference) ----
#define BATCH 128
#define TP    196            // patches per view
#define TWT   392            // 2*TP tokens
#define VIS   98             // visible tokens per sample
#define MSK   294            // masked tokens per sample
#define ENC   1024
#define DECD  512
#define MTOT  (BATCH * VIS)  // 12544 GEMM rows

// ---- GEMM tiling ----
#define BM 64                // block tile M
#define BN 128               // block tile N
#define BK 32                // k-chunk staged in LDS
#define LDA 36               // padded LDS stride (floats): conflict-free b64 reads
#define A_FLOATS (BM * LDA)          // 2304
#define B_FLOATS (BN * LDA)          // 4608
#define BUF_FLOATS (A_FLOATS + B_FLOATS)  // 6912 (27648 B per buffer)
#define CHUNKS (ENC / BK)            // 32
#define ASYNC_PER_CHUNK 12           // 4 (A) + 8 (B) b64 async copies / thread

typedef __attribute__((ext_vector_type(2))) float vf2;
typedef __attribute__((ext_vector_type(8))) float vf8;

// CDNA5 async copy: global -> LDS, tracked by ASYNCcnt (no VGPR data path).
__device__ __forceinline__ void async_copy_b64(unsigned lds_byte_addr,
                                               unsigned long long gaddr) {
  asm volatile("global_load_async_to_lds_b64 %0, %1, off"
               :: "v"(lds_byte_addr), "v"(gaddr)
               : "memory");
}

// ---------------------------------------------------------------------------
// Kernel A: visibility map.  map[b,t] = rank or -1 ; vis[b,r] = t
// ---------------------------------------------------------------------------
__global__ __launch_bounds__(128)
void build_map_kernel(const int* __restrict__ masked,
                      int* __restrict__ map, int* __restrict__ vis) {
  const int b = blockIdx.x;
  __shared__ int flags[TWT];
  for (int t = threadIdx.x; t < TWT; t += 128) flags[t] = 1;
  __syncthreads();
  for (int i = threadIdx.x; i < MSK; i += 128) flags[masked[b * MSK + i]] = 0;
  __syncthreads();
  for (int t = threadIdx.x; t < TWT; t += 128) {
    int r = 0;
    for (int s = 0; s < t; ++s) r += flags[s];
    if (flags[t]) {
      map[b * TWT + t] = r;
      vis[b * VIS + r] = t;
    } else {
      map[b * TWT + t] = -1;
    }
  }
}

// ---------------------------------------------------------------------------
// Kernel B: masked slots <- mask_token + pos_embed + view_embed (float4).
// ---------------------------------------------------------------------------
__global__ __launch_bounds__(256)
void fill_masked_kernel(const float* __restrict__ mask_token,
                        const float* __restrict__ pos,
                        const float* __restrict__ ve,
                        const int* __restrict__ map,
                        float* __restrict__ out) {
  const int idx = blockIdx.x * 256 + threadIdx.x;
  const int c  = idx & 127;
  const int bt = idx >> 7;
  if (bt >= BATCH * TWT) return;
  if (map[bt] >= 0) return;
  const int t = bt % TWT;
  float4 v = ((const float4*)mask_token)[c];
  float4 p = ((const float4*)(pos + (size_t)t * DECD))[c];
  float4 e = ((const float4*)(ve + (t < TP ? 0 : DECD)))[c];
  v.x += p.x + e.x; v.y += p.y + e.y; v.z += p.z + e.z; v.w += p.w + e.w;
  ((float4*)(out + (size_t)bt * DECD))[c] = v;
}

// ---------------------------------------------------------------------------
// Kernel C: fused GEMM + scatter, LDS double-buffered via async copies.
//   Block = 256 threads = 8 wave32 waves (4M x 2N), wave tile 16x64.
//   xp[m,n] = sum_k x[m,k]*W[n,k];  out[b, vis[b][r], n] = xp + bias+pos+ve
// ---------------------------------------------------------------------------
__global__ __launch_bounds__(256)
void gemm_scatter_kernel(const float* __restrict__ X,
                         const float* __restrict__ W,
                         const float* __restrict__ bias,
                         const float* __restrict__ pos,
                         const float* __restrict__ ve,
                         const int* __restrict__ vis,
                         float* __restrict__ out) {
  __shared__ float lds[2 * BUF_FLOATS];

  const int tid  = threadIdx.x;
  const int lane = tid & 31;
  const int wave = tid >> 5;
  const int wm = wave & 3;        // 0..3  (M waves)
  const int wn = wave >> 2;       // 0..1  (N waves)
  const int lo = lane & 15;
  const int hi = lane >> 4;

  const int m0b = blockIdx.y * BM;
  const int n0b = blockIdx.x * BN;
  const int m0 = m0b + wm * 16;
  const int n0 = n0b + wn * 64;

  // LDS aperture: low 32 bits of the generic address are the LDS offset.
  const unsigned lds_base = (unsigned)(size_t)(void*)lds;

  // ---- per-thread staging maps (coalesced: 16 consecutive lanes per row) ----
  unsigned      ldsAoff[4];  unsigned long long srcA[4];
#pragma unroll
  for (int i = 0; i < 4; ++i) {            // A: 64 rows x 32 cols
    const int j = tid + 256 * i;
    const int row = j >> 4, col2 = j & 15; // col2 = pair of floats
    srcA[i] = (unsigned long long)(X + (size_t)(m0b + row) * ENC + 2 * col2);
    ldsAoff[i] = (unsigned)(row * LDA + 2 * col2) * 4u;
  }
  unsigned      ldsBoff[8];  unsigned long long srcB[8];
#pragma unroll
  for (int i = 0; i < 8; ++i) {            // B: 128 rows x 32 cols
    const int j = tid + 256 * i;
    const int row = j >> 4, col2 = j & 15;
    srcB[i] = (unsigned long long)(W + (size_t)(n0b + row) * ENC + 2 * col2);
    ldsBoff[i] = (unsigned)(A_FLOATS * 4 + (row * LDA + 2 * col2) * 4) ;
  }

  vf8 acc0 = {}, acc1 = {}, acc2 = {}, acc3 = {};

  // ---- issue chunk 0 into buffer 0 ----
  {
    const unsigned base = lds_base;
#pragma unroll
    for (int i = 0; i < 4; ++i) async_copy_b64(base + ldsAoff[i], srcA[i]);
#pragma unroll
    for (int i = 0; i < 8; ++i) async_copy_b64(base + ldsBoff[i], srcB[i]);
#pragma unroll
    for (int i = 0; i < 4; ++i) srcA[i] += BK * 4;
#pragma unroll
    for (int i = 0; i < 8; ++i) srcB[i] += BK * 4;
  }

  for (int c = 0; c < CHUNKS; ++c) {
    if (c + 1 < CHUNKS) {
      // prefetch next chunk into the other buffer
      const unsigned base = lds_base + ((c + 1) & 1) * (BUF_FLOATS * 4);
#pragma unroll
      for (int i = 0; i < 4; ++i) async_copy_b64(base + ldsAoff[i], srcA[i]);
#pragma unroll
      for (int i = 0; i < 8; ++i) async_copy_b64(base + ldsBoff[i], srcB[i]);
#pragma unroll
      for (int i = 0; i < 4; ++i) srcA[i] += BK * 4;
#pragma unroll
      for (int i = 0; i < 8; ++i) srcB[i] += BK * 4;
      // async loads complete in order: <=12 pending => current chunk landed
      asm volatile("s_wait_asynccnt 12" ::: "memory");
    } else {
      asm volatile("s_wait_asynccnt 0" ::: "memory");
    }
    __syncthreads();   // all waves' copies of buf[c&1] complete

    const float* As = lds + (c & 1) * BUF_FLOATS;
    const float* Bs = As + A_FLOATS;
    // Fragment layout (f32 WMMA): lanes 0-15 = M/N index, VGPR0/1 = K{0,1}
    // (lo half) / K{2,3} (hi half). Stride 36 -> conflict-free ds_load_b64.
    const float* aP  = As + (wm * 16 + lo) * LDA + 2 * hi;
    const float* b0p = Bs + (wn * 64 +  0 + lo) * LDA + 2 * hi;
    const float* b1p = Bs + (wn * 64 + 16 + lo) * LDA + 2 * hi;
    const float* b2p = Bs + (wn * 64 + 32 + lo) * LDA + 2 * hi;
    const float* b3p = Bs + (wn * 64 + 48 + lo) * LDA + 2 * hi;

#pragma unroll
    for (int k = 0; k < BK; k += 4) {
      vf2 a  = *(const vf2*)(aP + k);
      vf2 b0 = *(const vf2*)(b0p + k);
      vf2 b1 = *(const vf2*)(b1p + k);
      vf2 b2 = *(const vf2*)(b2p + k);
      vf2 b3 = *(const vf2*)(b3p + k);
      acc0 = __builtin_amdgcn_wmma_f32_16x16x4_f32(false, a, false, b0,
                                                   (short)0, acc0, false, false);
      acc1 = __builtin_amdgcn_wmma_f32_16x16x4_f32(false, a, false, b1,
                                                   (short)0, acc1, false, false);
      acc2 = __builtin_amdgcn_wmma_f32_16x16x4_f32(false, a, false, b2,
                                                   (short)0, acc2, false, false);
      acc3 = __builtin_amdgcn_wmma_f32_16x16x4_f32(false, a, false, b3,
                                                   (short)0, acc3, false, false);
    }
    __syncthreads();   // everyone done reading buf[c&1] before it is refilled
  }

  // ---- epilogue: D row m = m0 + i + 8*hi, col n = n0 + 16*j + lo ----
  const int c0 = n0 + lo;
  const float bv0 = bias[c0],      bv1 = bias[c0 + 16];
  const float bv2 = bias[c0 + 32], bv3 = bias[c0 + 48];

#pragma unroll
  for (int i = 0; i < 8; ++i) {
    const int m = m0 + i + 8 * hi;
    const unsigned bidx = (unsigned)m / VIS;
    const unsigned r    = (unsigned)m - bidx * VIS;
    const int t = vis[bidx * VIS + r];
    const float* posr = pos + (size_t)t * DECD;
    const float* ver  = ve + (t < TP ? 0 : DECD);
    float* outr = out + ((size_t)bidx * TWT + (size_t)t) * DECD;
    outr[c0]      = acc0[i] + bv0 + posr[c0]      + ver[c0];
    outr[c0 + 16] = acc1[i] + bv1 + posr[c0 + 16] + ver[c0 + 16];
    outr[c0 + 32] = acc2[i] + bv2 + posr[c0 + 32] + ver[c0 + 32];
    outr[c0 + 48] = acc3[i] + bv3 + posr[c0 + 48] + ver[c0 + 48];
  }
}

// ---------------------------------------------------------------------------
extern "C" void kernel_launch(void* const* d_in, const int* in_sizes, int n_in,
                              void* d_out, int out_size, void* d_ws, size_t ws_size,
                              hipStream_t stream) {
  const float* x          = (const float*)d_in[0];  // (B, V, ENC)
  const int*   masked     = (const int*)d_in[1];    // (B, M)
  const float* W          = (const float*)d_in[2];  // (DEC, ENC)
  const float* bias       = (const float*)d_in[3];  // (DEC,)
  const float* mask_token = (const float*)d_in[4];  // (1,1,DEC)
  const float* pos        = (const float*)d_in[5];  // (1, 2T, DEC)
  const float* ve         = (const float*)d_in[6];  // (2, DEC)
  float* out = (float*)d_out;                       // (B, 2T, DEC)

  int* map = (int*)d_ws;
  int* vis = (int*)((char*)d_ws + (size_t)BATCH * TWT * sizeof(int));

  build_map_kernel<<<BATCH, 128, 0, stream>>>(masked, map, vis);

  const int fill_threads = BATCH * TWT * (DECD / 4);
  fill_masked_kernel<<<(fill_threads + 255) / 256, 256, 0, stream>>>(
      mask_token, pos, ve, map, out);

  dim3 grid(DECD / BN, MTOT / BM);   // (4, 196)
  gemm_scatter_kernel<<<grid, 256, 0, stream>>>(x, W, bias, pos, ve, vis, out);
}